// DoubleSAGEnoBN_49228915147576
// MI455X (gfx1250) — compile-verified
//
#include <hip/hip_runtime.h>

// ---------------------------------------------------------------------------
// DoubleSAGE (mean) + edge-MLP link predictor for MI455X (gfx1250, wave32).
//
// Math transform: mean_agg(x) @ W_neigh == segment_sum(x @ W_neigh) / deg,
// so we run the neighbor GEMM FIRST and scatter-add in the (smaller) output
// space: 1M x 256 floats instead of 1M x 512 (layer 1) and 1M x 64 instead of
// 1M x 256 (layer 2). The scatter is the HBM-bound critical path on this
// chip (23.3 TB/s), the fp32 GEMMs (~65 GFLOP total) are noise, so all GEMMs
// use V_WMMA_F32_16X16X4_F32 to stay in full fp32 precision.
// ---------------------------------------------------------------------------

typedef float v2f __attribute__((ext_vector_type(2)));
typedef float v8f __attribute__((ext_vector_type(8)));

__device__ __forceinline__ v8f wmma_f32_4(v2f a, v2f b, v8f c) {
  // (neg_a, A, neg_b, B, c_mod, C, reuse_a, reuse_b)
  return __builtin_amdgcn_wmma_f32_16x16x4_f32(false, a, false, b, (short)0, c,
                                               false, false);
}

// ---------------------------------------------------------------------------
// deg[v] += 1 for each edge v = dst[e]
// ---------------------------------------------------------------------------
__global__ void degree_kernel(const int* __restrict__ dst,
                              float* __restrict__ deg, int nE) {
  int gid = blockIdx.x * blockDim.x + threadIdx.x;
  if (gid < nE) unsafeAtomicAdd(&deg[dst[gid]], 1.0f);
}

// ---------------------------------------------------------------------------
// s[dst[e]][c] += y[src[e]][c]   (C = 1<<cshift columns per row)
// Consecutive lanes handle consecutive columns of one edge -> coalesced
// global_atomic_add_f32 traffic.
// ---------------------------------------------------------------------------
__global__ void scatter_add_kernel(const float* __restrict__ y,
                                   const int* __restrict__ src,
                                   const int* __restrict__ dst,
                                   float* __restrict__ s,
                                   int nE, int cshift) {
  const int gid = blockIdx.x * blockDim.x + threadIdx.x;
  const int e = gid >> cshift;
  const int c = gid & ((1 << cshift) - 1);
  if (e >= nE) return;
  const float val = y[((long)src[e] << cshift) + c];
  unsafeAtomicAdd(&s[((long)dst[e] << cshift) + c], val);
}

// ---------------------------------------------------------------------------
// C[M,N] = A[M,K] @ B[K,N]  (+ S/max(deg,1) + bias, optional ReLU)
// One wave32 owns one 16x16 C tile; K-loop in steps of 4 via
// V_WMMA_F32_16X16X4_F32.  A-frag per ISA 7.12.2: lanes 0-15 carry K={k,k+1},
// lanes 16-31 carry K={k+2,k+3} of rows M=lane&15 (8B vector loads).
// ---------------------------------------------------------------------------
__global__ void sage_gemm_wmma(const float* __restrict__ A,
                               const float* __restrict__ B,
                               float* __restrict__ C,
                               const float* __restrict__ S,
                               const float* __restrict__ deg,
                               const float* __restrict__ bias,
                               int M, int N, int K, int relu) {
  const int wave = blockIdx.x * (blockDim.x >> 5) + (threadIdx.x >> 5);
  const int tilesN = N >> 4;
  const int tm = wave / tilesN;
  const int tn = wave % tilesN;
  if (tm >= ((M + 15) >> 4)) return;  // wave-uniform: EXEC stays all-ones

  const int lane = threadIdx.x & 31;
  const int half = lane >> 4;
  const int l = lane & 15;

  int arow = tm * 16 + l;
  if (arow >= M) arow = M - 1;          // clamp loads (per-lane data op only)
  const int bcol = tn * 16 + l;
  const float* Arow = A + (long)arow * K;

  v8f acc = {};
  for (int k = 0; k < K; k += 4) {
    if ((k & 63) == 0) __builtin_prefetch(Arow + k + 64, 0, 1);
    const int kk = k + 2 * half;        // this half-wave's K pair
    const v2f a = *(const v2f*)(Arow + kk);
    v2f b;
    b.x = B[(long)kk * N + bcol];
    b.y = B[(long)(kk + 1) * N + bcol];
    acc = wmma_f32_4(a, b, acc);
  }

#pragma unroll
  for (int i = 0; i < 8; ++i) {
    const int row = tm * 16 + half * 8 + i;   // C layout: VGPR i -> row i(+8)
    if (row >= M) continue;
    float v = acc[i];
    if (S) {
      const float d = fmaxf(deg[row], 1.0f);
      v += S[(long)row * N + bcol] * (1.0f / d);
    }
    if (bias) v += bias[bcol];
    if (relu) v = fmaxf(v, 0.0f);
    C[(long)row * N + bcol] = v;
  }
}

// ---------------------------------------------------------------------------
// Edge scorer: per wave, 16 edges.  A = concat(h2[s], h2[d]) gathered on the
// fly as a 16x128 tile; 4 WMMA N-tiles against Wp1[128,64]; fused bp1+ReLU;
// then score = hidden . Wp2 + bp2 via in-register partials + shfl_xor tree.
// ---------------------------------------------------------------------------
__global__ void edge_score_wmma(const float* __restrict__ h2,  // [N,64]
                                const int* __restrict__ es,
                                const int* __restrict__ ed,
                                const float* __restrict__ Wp1,  // [128,64]
                                const float* __restrict__ bp1,  // [64]
                                const float* __restrict__ Wp2,  // [64]
                                const float* __restrict__ bp2,  // [1]
                                float* __restrict__ out, int E) {
  const int wave = blockIdx.x * (blockDim.x >> 5) + (threadIdx.x >> 5);
  const int base = wave * 16;
  if (base >= E) return;  // wave-uniform

  const int lane = threadIdx.x & 31;
  const int half = lane >> 4;
  const int l = lane & 15;

  int ei = base + l;
  if (ei >= E) ei = E - 1;  // clamp (data only, no EXEC divergence)
  const float* rowS = h2 + (long)es[ei] * 64;
  const float* rowD = h2 + (long)ed[ei] * 64;

  v8f a0 = {}, a1 = {}, a2 = {}, a3 = {};
  for (int k = 0; k < 128; k += 4) {
    const int kk = k + 2 * half;
    const float* arow = (kk < 64) ? (rowS + kk) : (rowD + kk - 64);
    const v2f a = *(const v2f*)arow;
    const float* B0 = Wp1 + (long)kk * 64;
    const float* B1 = Wp1 + (long)(kk + 1) * 64;
    v2f b;
    b.x = B0[l +  0]; b.y = B1[l +  0]; a0 = wmma_f32_4(a, b, a0);
    b.x = B0[l + 16]; b.y = B1[l + 16]; a1 = wmma_f32_4(a, b, a1);
    b.x = B0[l + 32]; b.y = B1[l + 32]; a2 = wmma_f32_4(a, b, a2);
    b.x = B0[l + 48]; b.y = B1[l + 48]; a3 = wmma_f32_4(a, b, a3);
  }

  const float bp2v = bp2[0];
  v8f accs[4] = {a0, a1, a2, a3};
  float part[8];
#pragma unroll
  for (int i = 0; i < 8; ++i) part[i] = 0.0f;
#pragma unroll
  for (int t = 0; t < 4; ++t) {
    const int col = t * 16 + l;
    const float w2 = Wp2[col];
    const float bb = bp1[col];
#pragma unroll
    for (int i = 0; i < 8; ++i) {
      const float hv = fmaxf(accs[t][i] + bb, 0.0f);
      part[i] += hv * w2;
    }
  }
  // Reduce the 16 lanes of each half-wave (each lane holds one column slice
  // of rows {half*8 .. half*8+7}).  xor masks 1..8 stay within a 16-lane half.
#pragma unroll
  for (int i = 0; i < 8; ++i) {
    float v = part[i];
    v += __shfl_xor(v, 1, 32);
    v += __shfl_xor(v, 2, 32);
    v += __shfl_xor(v, 4, 32);
    v += __shfl_xor(v, 8, 32);
    const int row = base + half * 8 + i;
    if (l == 0 && row < E) out[row] = v + bp2v;
  }
}

// ---------------------------------------------------------------------------
// Host orchestration (graph-capture safe: only kernel launches + memsetAsync)
// ---------------------------------------------------------------------------
static inline void launch_gemm(const float* A, const float* B, float* C,
                               const float* S, const float* deg,
                               const float* bias, int M, int N, int K,
                               int relu, hipStream_t stream) {
  const int tiles = ((M + 15) / 16) * (N / 16);
  const int blocks = (tiles + 7) / 8;  // 8 waves (256 threads) per WG
  sage_gemm_wmma<<<blocks, 256, 0, stream>>>(A, B, C, S, deg, bias, M, N, K,
                                             relu);
}

extern "C" void kernel_launch(void* const* d_in, const int* in_sizes, int n_in,
                              void* d_out, int out_size, void* d_ws,
                              size_t ws_size, hipStream_t stream) {
  const float* x        = (const float*)d_in[0];
  const int*   src      = (const int*)d_in[1];
  const int*   dst      = (const int*)d_in[2];
  const int*   pos_src  = (const int*)d_in[3];
  const int*   pos_dst  = (const int*)d_in[4];
  const int*   neg_src  = (const int*)d_in[5];
  const int*   neg_dst  = (const int*)d_in[6];
  const float* W_self1  = (const float*)d_in[7];
  const float* W_neigh1 = (const float*)d_in[8];
  const float* b1       = (const float*)d_in[9];
  const float* W_self2  = (const float*)d_in[10];
  const float* W_neigh2 = (const float*)d_in[11];
  const float* b2       = (const float*)d_in[12];
  const float* Wp1      = (const float*)d_in[13];
  const float* bp1      = (const float*)d_in[14];
  const float* Wp2      = (const float*)d_in[15];
  const float* bp2      = (const float*)d_in[16];
  float* out = (float*)d_out;

  const int NN = 100000;
  const int E  = in_sizes[1];   // 1,000,000
  const int EP = in_sizes[3];   // 200,000
  const int D_IN = 512, D_HID = 256, D_OUT = 64;

  // Workspace layout (256B aligned): deg | bufA(y1->h1) | s1 | bufC(y2->h2) | s2
  char* ws = (char*)d_ws;
  size_t off = 0;
  auto alloc = [&](size_t bytes) {
    void* p = ws + off;
    off += (bytes + 255) & ~(size_t)255;
    return p;
  };
  float* deg  = (float*)alloc((size_t)NN * 4);
  float* bufA = (float*)alloc((size_t)NN * D_HID * 4);  // y1, then h1
  float* s1   = (float*)alloc((size_t)NN * D_HID * 4);
  float* bufC = (float*)alloc((size_t)NN * D_OUT * 4);  // y2, then h2
  float* s2   = (float*)alloc((size_t)NN * D_OUT * 4);
  (void)ws_size;

  // Atomic targets must be zeroed every call (graph replays).
  hipMemsetAsync(deg, 0, (size_t)NN * 4, stream);
  hipMemsetAsync(s1, 0, (size_t)NN * D_HID * 4, stream);
  hipMemsetAsync(s2, 0, (size_t)NN * D_OUT * 4, stream);

  // In-degrees.
  degree_kernel<<<(E + 255) / 256, 256, 0, stream>>>(dst, deg, E);

  // --- Layer 1 (aggregate-after-GEMM) ---
  // y1 = x @ W_neigh1                       [NN,256]
  launch_gemm(x, W_neigh1, bufA, nullptr, nullptr, nullptr, NN, D_HID, D_IN, 0,
              stream);
  // s1[dst] += y1[src]                      (1M x 256 scatter; the HBM wall)
  scatter_add_kernel<<<(E * (D_HID / 256)) * 1, 256, 0, stream>>>(
      bufA, src, dst, s1, E, 8);
  // h1 = relu(x @ W_self1 + s1/deg + b1)    [NN,256] (overwrites y1)
  launch_gemm(x, W_self1, bufA, s1, deg, b1, NN, D_HID, D_IN, 1, stream);

  // --- Layer 2 ---
  // y2 = h1 @ W_neigh2                      [NN,64]
  launch_gemm(bufA, W_neigh2, bufC, nullptr, nullptr, nullptr, NN, D_OUT,
              D_HID, 0, stream);
  // s2[dst] += y2[src]                      (1M x 64 scatter)
  scatter_add_kernel<<<(E * D_OUT) / 256, 256, 0, stream>>>(bufC, src, dst, s2,
                                                            E, 6);
  // h2 = h1 @ W_self2 + s2/deg + b2         [NN,64] (overwrites y2)
  launch_gemm(bufA, W_self2, bufC, s2, deg, b2, NN, D_OUT, D_HID, 0, stream);

  // --- Edge scores: 16 edges per wave, 8 waves per WG ---
  const int waves = (EP + 15) / 16;
  const int blocks = (waves + 7) / 8;
  edge_score_wmma<<<blocks, 256, 0, stream>>>(bufC, pos_src, pos_dst, Wp1, bp1,
                                              Wp2, bp2, out, EP);
  edge_score_wmma<<<blocks, 256, 0, stream>>>(bufC, neg_src, neg_dst, Wp1, bp1,
                                              Wp2, bp2, out + EP, EP);
}